// OneDimEquivalent_10977936409118
// MI455X (gfx1250) — compile-verified
//
#include <hip/hip_runtime.h>
#include <hip/hip_bf16.h>
#include <stdint.h>

// ---------------------------------------------------------------------------
// OneDimEquivalent scan for MI455X (gfx1250).
//
// Pipeline (all on `stream`, sequential):
//   K1 gl_setup   : fp64 Newton for the 32 positive roots/weights of P_64
//                   (matches numpy leggauss to fp64 precision).
//   K2 lut_build  : G(s) = sum_j c_j*sech^2(sqrt(s)*z_j) for 6145 grid points,
//                   reduction done with chained V_WMMA_F32_16X16X4_F32.
//   K3 lut_pack   : (value, slope) float2 pairs; writes out[0]=0.
//   K4 scan       : single wave32; TDM tensor_load_to_lds stages LUT into LDS;
//                   ~60-cycle serial chain per step, u prefetched 32 steps
//                   ahead in registers and ~256 steps ahead into GL2.
//
// Workspace layout (bytes), ~75 KB total:
//   [0     ] double zA[32]      (A * positive GL nodes)
//   [256   ] double cc[32]      (2*A*w_j*exp(-zA^2/2)/(2pi))
//   [512   ] float  val[6145]
//   [25600 ] float2 pair[6144]
// ---------------------------------------------------------------------------

#define GL_N    64
#define NLUT    6144
#define NVAL    (NLUT + 1)
#define DS_D    0.0078125          // 48 / 6144, exact in binary
#define INV_DS_F 128.0f

#define WS_ZA   0
#define WS_CC   256
#define WS_VAL  512
#define WS_PAIR 25600

typedef float v2f __attribute__((ext_vector_type(2)));
typedef float v8f __attribute__((ext_vector_type(8)));

// ---------------------------------------------------------------------------
// K1: 64-pt Gauss-Legendre positive nodes & folded coefficients (fp64 Newton).
// ---------------------------------------------------------------------------
__global__ __launch_bounds__(32) void gl_setup_k(double* zA, double* cc) {
    int i = threadIdx.x;
    if (i >= 32) return;
    const double PI = 3.14159265358979323846264338327950288;
    const int n = GL_N;
    // k-th root (k = i+1), descending: first 32 are the positive roots.
    double x = cos(PI * ((double)i + 0.75) / ((double)n + 0.5));
    double dp = 1.0;
    for (int it = 0; it < 40; ++it) {
        double p0 = 1.0, p1 = x;
        for (int j = 2; j <= n; ++j) {
            double p2 = ((2.0 * j - 1.0) * x * p1 - (j - 1.0) * p0) / (double)j;
            p0 = p1; p1 = p2;
        }
        dp = (double)n * (x * p1 - p0) / (x * x - 1.0);
        x -= p1 / dp;
    }
    { // recompute derivative at converged root for the weight
        double p0 = 1.0, p1 = x;
        for (int j = 2; j <= n; ++j) {
            double p2 = ((2.0 * j - 1.0) * x * p1 - (j - 1.0) * p0) / (double)j;
            p0 = p1; p1 = p2;
        }
        dp = (double)n * (x * p1 - p0) / (x * x - 1.0);
    }
    double w = 2.0 / ((1.0 - x * x) * dp * dp);
    const double A = 5.0;
    const double TWO_PI = 6.28318530717958647692528676655900577;
    double z = A * x;
    zA[i] = z;
    // even integrand: 2x the positive-node sum; fold A*w, gaussian, 1/(2pi)
    cc[i] = 2.0 * (A * w) * exp(-0.5 * z * z) / TWO_PI;
}

// ---------------------------------------------------------------------------
// K2: LUT of G(s). One wave computes 16 grid points; the 32-node reduction is
// 8 chained V_WMMA_F32_16X16X4_F32 with coefficients folded into the A tile
// (documented 16x4 fp32 A layout) and B = all-ones (layout-independent).
// D[m][n] = sum_k A[m][k] identical for every n -> read column n=0.
// ---------------------------------------------------------------------------
__device__ __forceinline__ float sech2f(float y) {  // y >= 0
    float e = __expf(-2.0f * y);
    float d = 1.0f + e;
#if defined(__has_builtin)
#if __has_builtin(__builtin_amdgcn_rcpf)
    float r = __builtin_amdgcn_rcpf(d);   // v_rcp_f32, ~1 ulp: fine for the LUT
#else
    float r = 1.0f / d;
#endif
#else
    float r = 1.0f / d;
#endif
    return 4.0f * e * r * r;
}

__global__ __launch_bounds__(32) void lut_build_k(const double* __restrict__ zA,
                                                  const double* __restrict__ cc,
                                                  float* __restrict__ val) {
    int lane = threadIdx.x;
    int base = blockIdx.x * 16;
    int m    = lane & 15;          // row index of A / D
    int hi   = lane >> 4;          // lanes 16..31 supply K=2,3 of each group
    float s     = (float)(base + m) * (float)DS_D;
    float delta = sqrtf(s);

    v8f acc = {0.f, 0.f, 0.f, 0.f, 0.f, 0.f, 0.f, 0.f};
    v2f bone; bone.x = 1.0f; bone.y = 1.0f;

#pragma unroll
    for (int g = 0; g < 8; ++g) {
        int kx = 4 * g + (hi ? 2 : 0);   // A layout: VGPR0 lanes0-15 K=0, lanes16-31 K=2
        int ky = kx + 1;                 //           VGPR1 lanes0-15 K=1, lanes16-31 K=3
        float a0 = (float)cc[kx] * sech2f(delta * (float)zA[kx]);
        float a1 = (float)cc[ky] * sech2f(delta * (float)zA[ky]);
        v2f a; a.x = a0; a.y = a1;
        acc = __builtin_amdgcn_wmma_f32_16x16x4_f32(
            /*neg_a=*/false, a, /*neg_b=*/false, bone,
            /*c_mod=*/(short)0, acc, /*reuse_a=*/false, /*reuse_b=*/false);
    }
    // D layout: VGPR r, lanes0-15 -> M=r (N=lane); lanes16-31 -> M=8+r.
    if (lane == 0) {
#pragma unroll
        for (int r = 0; r < 8; ++r)
            if (base + r < NVAL) val[base + r] = acc[r];
    } else if (lane == 16) {
#pragma unroll
        for (int r = 0; r < 8; ++r)
            if (base + 8 + r < NVAL) val[base + 8 + r] = acc[r];
    }
}

// ---------------------------------------------------------------------------
// K3: pack (value, slope) pairs for single-ds_load_b64 interpolation.
// ---------------------------------------------------------------------------
__global__ void lut_pack_k(const float* __restrict__ val,
                           float2* __restrict__ pair,
                           float* __restrict__ out) {
    int i = blockIdx.x * blockDim.x + threadIdx.x;
    if (i == 0) out[0] = 0.0f;                       // z_hist[0] = 0
    if (i < NLUT) pair[i] = make_float2(val[i], val[i + 1] - val[i]);
}

// ---------------------------------------------------------------------------
// K4: the sequential scan. Single wave32. LUT in LDS (broadcast reads, no bank
// conflicts). u prefetched one 32-step block ahead; z written coalesced.
// ---------------------------------------------------------------------------
__global__ __launch_bounds__(32) void scan_k(const float* __restrict__ u,
                                             const float2* __restrict__ pair,
                                             float* __restrict__ out, int T) {
    __shared__ float2 lut[NLUT];
    const int lane = threadIdx.x;

    // --- stage LUT into LDS: TDM async-tensor path (gfx1250), then a plain
    // cooperative copy as correctness backstop (overwrites same region).
#if defined(__has_builtin)
#if __has_builtin(__builtin_amdgcn_tensor_load_to_lds)
    {
        typedef unsigned int u32x4 __attribute__((ext_vector_type(4)));
        typedef int i32x4 __attribute__((ext_vector_type(4)));
        typedef int i32x8 __attribute__((ext_vector_type(8)));
        uint32_t lds_off = (uint32_t)(uintptr_t)(void*)&lut[0];
        uint64_t ga = (uint64_t)(uintptr_t)pair;
        const uint32_t DW = (uint32_t)NLUT * 2u;     // 12288 dwords = 48KB
        u32x4 g0;
        g0[0] = 1u;                                   // count=1 valid descriptor
        g0[1] = lds_off;                              // lds_addr
        g0[2] = (uint32_t)ga;                         // global_addr[31:0]
        g0[3] = ((uint32_t)(ga >> 32) & 0x01FFFFFFu)  // global_addr[56:32]
                | 0x80000000u;                        // type=2 ("image")
        i32x8 g1;
        g1[0] = 0x00020000;                           // wg_mask=0, data_size=4B
        g1[1] = (int)(DW << 16);                      // tensor_dim0[15:0]
        g1[2] = 0x00010000;                           // tensor_dim1 = 1
        g1[3] = (int)(DW << 16);                      // tile_dim0
        g1[4] = 0x00000001;                           // tile_dim1=1, tile_dim2=0
        g1[5] = (int)DW;                              // tensor_dim0_stride[31:0]
        g1[6] = (int)(DW << 16);                      // tensor_dim1_stride[15:0]
        g1[7] = 0;
        i32x4 gz = {0, 0, 0, 0};
#if defined(__clang_major__) && (__clang_major__ >= 23)
#pragma message("CDNA5-DIAG: TDM path = 6-arg __builtin_amdgcn_tensor_load_to_lds (clang>=23)")
        i32x8 gz8 = {0, 0, 0, 0, 0, 0, 0, 0};
        __builtin_amdgcn_tensor_load_to_lds(g0, g1, gz, gz, gz8, 0);
#else
#pragma message("CDNA5-DIAG: TDM path = 5-arg __builtin_amdgcn_tensor_load_to_lds (clang<23)")
        __builtin_amdgcn_tensor_load_to_lds(g0, g1, gz, gz, 0);
#endif
        __builtin_amdgcn_s_wait_tensorcnt((short)0);
    }
#else
#pragma message("CDNA5-DIAG: TDM builtin NOT available on this target/pass")
#endif
#endif
    for (int i = lane; i < NLUT; i += 32) lut[i] = pair[i];
    __syncthreads();

    float k = 0.0f, v = 0.0f, zsave = 0.0f;

    // one recurrence step; serial chain: mul,fma,mul,cvt,min,ds_load_b64,fma,fma,fma
    auto step = [&](float uu) -> float {
        float su2  = 0.64f * uu * uu;                 // off-chain (u known early)
        float s    = fmaf(1.44f, k * k, su2);
        float tt   = s * INV_DS_F;
        int   idx  = (int)tt;
        idx        = idx < (NLUT - 1) ? idx : (NLUT - 1);
        float frac = fminf(tt - (float)idx, 1.0f);    // parallel with LDS load
        float2 e   = lut[idx];                        // uniform addr -> broadcast
        float gi   = fmaf(e.y, frac, e.x);
        float aa   = fmaf(0.18f, gi, 0.8f);           // k*(0.8+0.18*gi)+0.1*gi*v
        float bb   = gi * (0.1f * v);
        float kn   = fmaf(k, aa, bb);
        float z    = 0.7f * gi * kn;                  // sig_mw_hat * k_new
        v = fmaf(0.2f, uu - v, v);                    // v += 0.2*(u - v)
        k = kn;
        return z;
    };
    auto load_block = [&](int b, float* buf) {        // 32 uniform floats
        const float4* p = (const float4*)(u + (b << 5));
#pragma unroll
        for (int q = 0; q < 8; ++q) {
            float4 t4 = p[q];
            buf[4 * q + 0] = t4.x; buf[4 * q + 1] = t4.y;
            buf[4 * q + 2] = t4.z; buf[4 * q + 3] = t4.w;
        }
    };

    float bufA[32], bufB[32];
    const int nb = T >> 5;
    if (nb > 0) load_block(0, bufA);
    for (int blk = 0; blk < nb; blk += 2) {
        if (blk + 8 < nb)                              // warm GL2 ~256 steps ahead
            __builtin_prefetch(u + ((blk + 8) << 5), 0, 1);
        if (blk + 1 < nb) load_block(blk + 1, bufB);  // prefetch next block
#pragma unroll
        for (int j = 0; j < 32; ++j) {
            float z = step(bufA[j]);
            zsave = (j == lane) ? z : zsave;          // lane-rotate the outputs
        }
        out[1 + (blk << 5) + lane] = zsave;           // coalesced b32 store
        if (blk + 2 < nb) load_block(blk + 2, bufA);  // prefetch block after
        if (blk + 1 < nb) {
#pragma unroll
            for (int j = 0; j < 32; ++j) {
                float z = step(bufB[j]);
                zsave = (j == lane) ? z : zsave;
            }
            out[1 + ((blk + 1) << 5) + lane] = zsave;
        }
    }
    for (int t = nb << 5; t < T; ++t) {               // tail (T%32 != 0)
        float z = step(u[t]);
        if (lane == 0) out[1 + t] = z;
    }
}

// ---------------------------------------------------------------------------
extern "C" void kernel_launch(void* const* d_in, const int* in_sizes, int n_in,
                              void* d_out, int out_size, void* d_ws, size_t ws_size,
                              hipStream_t stream) {
    const float* u = (const float*)d_in[0];
    int T = in_sizes[0];
    float* out = (float*)d_out;
    char* ws = (char*)d_ws;

    double* zA   = (double*)(ws + WS_ZA);
    double* cc   = (double*)(ws + WS_CC);
    float*  val  = (float*)(ws + WS_VAL);
    float2* pair = (float2*)(ws + WS_PAIR);

    gl_setup_k<<<dim3(1), dim3(32), 0, stream>>>(zA, cc);
    lut_build_k<<<dim3((NVAL + 15) / 16), dim3(32), 0, stream>>>(zA, cc, val);
    lut_pack_k<<<dim3((NLUT + 255) / 256), dim3(256), 0, stream>>>(val, pair, out);
    scan_k<<<dim3(1), dim3(32), 0, stream>>>(u, pair, out, T);
}